// SFEConv_with_global_pool_19121194402170
// MI455X (gfx1250) — compile-verified
//
#include <hip/hip_runtime.h>
#include <math.h>

#define HID 128
#define HEAD 8
#define DK 16
#define FN 36
#define FE 10
#define INV_SQRT_HID 0.08838834764831845f

typedef float v2f __attribute__((ext_vector_type(2)));
typedef float v8f __attribute__((ext_vector_type(8)));

__device__ __forceinline__ v8f wmma4(v2f a, v2f b, v8f c) {
  // V_WMMA_F32_16X16X4_F32 : D = A(16x4) x B(4x16) + C(16x16), all f32
  return __builtin_amdgcn_wmma_f32_16x16x4_f32(false, a, false, b, (short)0, c,
                                               false, false);
}

// ---------------- prep kernels ----------------

__global__ void k_h0(const float* __restrict__ x, float* __restrict__ h, int N) {
  int n = blockIdx.x * blockDim.x + threadIdx.x;
  if (n >= N) return;
  float ss = 0.f;
  for (int c = 0; c < FN; ++c) { float v = x[n * FN + c]; ss += v * v; }
  float inv = 1.f / fmaxf(sqrtf(ss), 1e-12f);
  for (int c = 0; c < HID; ++c)
    h[(size_t)n * HID + c] = (c < FN) ? x[n * FN + c] * inv : 0.f;
}

__global__ void k_ea(const float* __restrict__ ea, float* __restrict__ ean, int E) {
  int e = blockIdx.x * blockDim.x + threadIdx.x;
  if (e >= E) return;
  float ss = 0.f;
  for (int c = 0; c < FE; ++c) { float v = ea[e * FE + c]; ss += v * v; }
  float inv = 1.f / fmaxf(sqrtf(ss), 1e-12f);
  for (int c = 0; c < FE; ++c) ean[e * FE + c] = ea[e * FE + c] * inv;
}

__global__ void k_pad_we(const float* __restrict__ W, float* __restrict__ Wp) {
  int i = blockIdx.x * blockDim.x + threadIdx.x;   // (10,36) -> (10,128)
  if (i >= FE * HID) return;
  int r = i / HID, c = i % HID;
  Wp[i] = (c < FN) ? W[r * FN + c] : 0.f;
}

__global__ void k_pad_w128(const float* __restrict__ W, float* __restrict__ Wp) {
  int i = blockIdx.x * blockDim.x + threadIdx.x;   // (36,128) -> (128,128)
  if (i >= HID * HID) return;
  int r = i / HID, c = i % HID;
  Wp[i] = (r < FN) ? W[r * HID + c] : 0.f;
}

__global__ void k_fold_wk(const float* __restrict__ Wk, float* __restrict__ Wkf,
                          int Din) {                // [Din,128] -> [128,16]
  int i = blockIdx.x * blockDim.x + threadIdx.x;
  if (i >= HID * DK) return;
  int k = i / DK, d = i % DK;
  float s = 0.f;
  if (k < Din)
    for (int h = 0; h < HEAD; ++h) s += Wk[k * HID + h * DK + d];
  Wkf[k * DK + d] = s;
}

// ---------------- CSR build (deterministic) ----------------

__global__ void k_zero_i(int* __restrict__ p, int n) {
  int i = blockIdx.x * blockDim.x + threadIdx.x;
  if (i < n) p[i] = 0;
}
__global__ void k_count(const int* __restrict__ dst, int* __restrict__ deg, int E) {
  int e = blockIdx.x * blockDim.x + threadIdx.x;
  if (e < E) atomicAdd(&deg[dst[e]], 1);
}
__global__ void k_scan(const int* __restrict__ deg, int* __restrict__ off, int N) {
  if (blockIdx.x == 0 && threadIdx.x == 0) {
    int a = 0;
    for (int i = 0; i < N; ++i) { off[i] = a; a += deg[i]; }
    off[N] = a;
  }
}
__global__ void k_fill(const int* __restrict__ dst, const int* __restrict__ off,
                       int* __restrict__ cur, int* __restrict__ eid, int E) {
  int e = blockIdx.x * blockDim.x + threadIdx.x;
  if (e >= E) return;
  int d = dst[e];
  int p = atomicAdd(&cur[d], 1);
  eid[off[d] + p] = e;
}
__global__ void k_sort_csr(const int* __restrict__ off, int* __restrict__ eid, int N) {
  int n = blockIdx.x * blockDim.x + threadIdx.x;
  if (n >= N) return;
  int b = off[n], e = off[n + 1];
  for (int i = b + 1; i < e; ++i) {
    int v = eid[i], j = i - 1;
    while (j >= b && eid[j] > v) { eid[j + 1] = eid[j]; --j; }
    eid[j + 1] = v;
  }
}

// ---------------- WMMA node GEMM: C[M,Nc] = A[M,K] @ B[K,Nc] (+bias) ----------------
// one wave per 16x16 C-tile; gridDim = (M/16, Nc/16), blockDim = 32

__global__ void k_gemm_node(const float* __restrict__ A, const float* __restrict__ B,
                            const float* __restrict__ bias, float* __restrict__ C,
                            int M, int K, int Nc) {
  int m0 = blockIdx.x * 16;
  int n0 = blockIdx.y * 16;
  int lane = threadIdx.x & 31;
  int half = lane >> 4, id = lane & 15;
  int arow = min(m0 + id, M - 1);
  v8f acc = {};
  for (int k = 0; k < K; k += 4) {
    v2f a, b;
    const float* ap = A + (size_t)arow * K + k + 2 * half;
    a[0] = ap[0];
    a[1] = ap[1];
    const float* bp = B + (size_t)(k + 2 * half) * Nc + n0 + id;
    b[0] = bp[0];
    b[1] = bp[Nc];
    acc = wmma4(a, b, acc);
  }
  float bv = bias ? bias[n0 + id] : 0.f;
  for (int r = 0; r < 8; ++r) {
    int row = m0 + r + 8 * half;
    if (row < M) C[(size_t)row * Nc + n0 + id] = acc[r] + bv;
  }
}

// ---------------- edge kernel ----------------
// 16 edges per block, 256 threads (8 waves). Wave w handles output cols
// [16w,16w+16) == head w. emb = l2norm(ea@We); q = XQ[src]+emb@Wq;
// v = XV[src]+emb@Wv; energy[e][w] = <q_head_w, ksum[dst]>/sqrt(HID).

__global__ void k_edge(const float* __restrict__ ean, const int* __restrict__ src,
                       const int* __restrict__ dst, const float* __restrict__ We,
                       const float* __restrict__ Wq, const float* __restrict__ Wv,
                       const float* __restrict__ XQ, const float* __restrict__ XV,
                       const float* __restrict__ ksum, float* __restrict__ v_e,
                       float* __restrict__ energy, int E) {
  __shared__ float s_ea[16][FE];
  __shared__ float s_emb[16][HID + 2];
  __shared__ float s_inv[16];
  __shared__ float s_q[8][16][DK + 1];
  int e0 = blockIdx.x * 16;
  int t = threadIdx.x;

  if (t < 16 * FE) {
    int r = t / FE, c = t % FE;
    int e = min(e0 + r, E - 1);
    s_ea[r][c] = ean[(size_t)e * FE + c];
  }
  __syncthreads();

  for (int i = t; i < 16 * HID; i += 256) {
    int r = i >> 7, c = i & (HID - 1);
    float s = 0.f;
#pragma unroll
    for (int j = 0; j < FE; ++j) s += s_ea[r][j] * We[j * HID + c];
    s_emb[r][c] = s;
  }
  __syncthreads();
  if (t < 16) {
    float ss = 0.f;
    for (int c = 0; c < HID; ++c) { float v = s_emb[t][c]; ss += v * v; }
    s_inv[t] = 1.f / fmaxf(sqrtf(ss), 1e-12f);
  }
  __syncthreads();
  for (int i = t; i < 16 * HID; i += 256) {
    int r = i >> 7;
    s_emb[r][i & (HID - 1)] *= s_inv[r];
  }
  __syncthreads();

  int w = t >> 5, lane = t & 31, half = lane >> 4, id = lane & 15;
  int n0 = w * 16;
  v8f q = {}, v = {};
  for (int k = 0; k < HID; k += 4) {
    v2f a, bq, bv;
    a[0] = s_emb[id][k + 2 * half];
    a[1] = s_emb[id][k + 2 * half + 1];
    bq[0] = Wq[(k + 2 * half) * HID + n0 + id];
    bq[1] = Wq[(k + 2 * half + 1) * HID + n0 + id];
    q = wmma4(a, bq, q);
    bv[0] = Wv[(k + 2 * half) * HID + n0 + id];
    bv[1] = Wv[(k + 2 * half + 1) * HID + n0 + id];
    v = wmma4(a, bv, v);
  }
  for (int r = 0; r < 8; ++r) {
    int m = r + 8 * half;
    int e = e0 + m;
    if (e < E) {
      int sn = src[e];
      float qq = q[r] + XQ[(size_t)sn * HID + n0 + id];
      float vv = v[r] + XV[(size_t)sn * HID + n0 + id];
      v_e[(size_t)e * HID + n0 + id] = vv;
      s_q[w][m][id] = qq;
    }
  }
  __syncthreads();
  if (lane < 16) {
    int e = e0 + lane;
    if (e < E) {
      int dn = dst[e];
      float s = 0.f;
#pragma unroll
      for (int d = 0; d < DK; ++d) s += s_q[w][lane][d] * ksum[(size_t)dn * DK + d];
      energy[(size_t)e * HEAD + w] = s * INV_SQRT_HID;
    }
  }
}

// ---------------- per-node softmax + aggregation (incl. self loop) ----------------

__global__ void k_agg(const float* __restrict__ XQ, const float* __restrict__ XV,
                      const float* __restrict__ ksum, const float* __restrict__ energy,
                      const float* __restrict__ v_e, const int* __restrict__ off,
                      const int* __restrict__ eid, float* __restrict__ agg, int N) {
  int n = blockIdx.x;
  int t = threadIdx.x;  // 128
  __shared__ float s_m[HEAD], s_s[HEAD], s_self[HEAD];
  int beg = off[n], end = off[n + 1];
  if (t < HEAD) {
    float es = 0.f;
#pragma unroll
    for (int d = 0; d < DK; ++d)
      es += XQ[(size_t)n * HID + t * DK + d] * ksum[(size_t)n * DK + d];
    es *= INV_SQRT_HID;
    float m = es;
    for (int i = beg; i < end; ++i)
      m = fmaxf(m, energy[(size_t)eid[i] * HEAD + t]);
    float s = expf(es - m);
    for (int i = beg; i < end; ++i)
      s += expf(energy[(size_t)eid[i] * HEAD + t] - m);
    s_m[t] = m; s_s[t] = s; s_self[t] = es;
  }
  __syncthreads();
  int h = t >> 4;
  float m = s_m[h], inv_s = 1.f / s_s[h];
  float acc = expf(s_self[h] - m) * inv_s * XV[(size_t)n * HID + t];
  for (int i = beg; i < end; ++i) {
    int e = eid[i];
    float a = expf(energy[(size_t)e * HEAD + h] - m) * inv_s;
    acc += a * v_e[(size_t)e * HID + t];
  }
  agg[(size_t)n * HID + t] = acc;
}

// ---------------- double LayerNorm + tanh ----------------

__global__ void k_ln_tanh(const float* __restrict__ z, const float* __restrict__ g,
                          const float* __restrict__ b, float* __restrict__ out, int N) {
  int n = blockIdx.x, t = threadIdx.x;  // 128
  __shared__ float red[HID];
  float v = z[(size_t)n * HID + t];
  for (int pass = 0; pass < 2; ++pass) {
    red[t] = v;
    __syncthreads();
    for (int s = HID / 2; s > 0; s >>= 1) {
      if (t < s) red[t] += red[t + s];
      __syncthreads();
    }
    float mean = red[0] / (float)HID;
    __syncthreads();
    float d = v - mean;
    red[t] = d * d;
    __syncthreads();
    for (int s = HID / 2; s > 0; s >>= 1) {
      if (t < s) red[t] += red[t + s];
      __syncthreads();
    }
    float var = red[0] / (float)HID;
    __syncthreads();
    v = d * rsqrtf(var + 1e-5f) * g[t] + b[t];
  }
  out[(size_t)n * HID + t] = tanhf(v);
}

// ---------------- per-graph attention pooling + MLP head ----------------

__global__ void k_pool(const float* __restrict__ h, const int* __restrict__ ipos,
                       const float* __restrict__ gate, const float* __restrict__ W1,
                       const float* __restrict__ b1, const float* __restrict__ W2,
                       const float* __restrict__ b2, float* __restrict__ out, int P) {
  int g = blockIdx.x, t = threadIdx.x;  // 128 threads, P <= 128
  __shared__ float s_sc[128];
  __shared__ int s_node[128];
  __shared__ float s_pool[HID];
  __shared__ float s_o1[HID / 2];
  __shared__ float red[HID];
  if (t < P) {
    int node = ipos[g * P + t];
    s_node[t] = node;
    float s = 0.f;
    for (int c = 0; c < HID; ++c) s += h[(size_t)node * HID + c] * gate[c];
    s_sc[t] = s;
  }
  __syncthreads();
  if (t == 0) {
    float m = -3.4e38f;
    for (int i = 0; i < P; ++i) m = fmaxf(m, s_sc[i]);
    float ssum = 0.f;
    for (int i = 0; i < P; ++i) { float e = expf(s_sc[i] - m); s_sc[i] = e; ssum += e; }
    float inv = 1.f / ssum;
    for (int i = 0; i < P; ++i) s_sc[i] *= inv;
  }
  __syncthreads();
  {
    float acc = 0.f;
    for (int i = 0; i < P; ++i) acc += s_sc[i] * h[(size_t)s_node[i] * HID + t];
    s_pool[t] = acc;
  }
  __syncthreads();
  if (t < HID / 2) {
    float acc = b1[t];
    for (int f = 0; f < HID; ++f) acc += s_pool[f] * W1[f * (HID / 2) + t];
    s_o1[t] = tanhf(acc);
  }
  __syncthreads();
  red[t] = (t < HID / 2) ? s_o1[t] * W2[t] : 0.f;
  __syncthreads();
  for (int s = HID / 2; s > 0; s >>= 1) {
    if (t < s) red[t] += red[t + s];
    __syncthreads();
  }
  if (t == 0) out[g] = 1.f / (1.f + expf(-(red[0] + b2[0])));
}

// ---------------- host launcher ----------------

extern "C" void kernel_launch(void* const* d_in, const int* in_sizes, int n_in,
                              void* d_out, int out_size, void* d_ws, size_t ws_size,
                              hipStream_t stream) {
  (void)n_in; (void)ws_size;
  const float* x        = (const float*)d_in[0];
  const float* edge_attr= (const float*)d_in[1];
  const float* c0_We = (const float*)d_in[2];
  const float* c0_Wq = (const float*)d_in[3];
  const float* c0_Wk = (const float*)d_in[4];
  const float* c0_Wv = (const float*)d_in[5];
  const float* c0_Wo = (const float*)d_in[6];
  const float* c0_bo = (const float*)d_in[7];
  const float* c0_g  = (const float*)d_in[8];
  const float* c0_b  = (const float*)d_in[9];
  const float* cs_We = (const float*)d_in[10];
  const float* cs_Wq = (const float*)d_in[11];
  const float* cs_Wk = (const float*)d_in[12];
  const float* cs_Wv = (const float*)d_in[13];
  const float* cs_Wo = (const float*)d_in[14];
  const float* cs_bo = (const float*)d_in[15];
  const float* cs_g  = (const float*)d_in[16];
  const float* cs_b  = (const float*)d_in[17];
  const float* gate  = (const float*)d_in[18];
  const float* W1    = (const float*)d_in[19];
  const float* b1    = (const float*)d_in[20];
  const float* W2    = (const float*)d_in[21];
  const float* b2    = (const float*)d_in[22];
  const int*   eidx  = (const int*)d_in[23];
  const int*   ipos  = (const int*)d_in[25];

  const int N = in_sizes[0] / FN;
  const int E = in_sizes[1] / FE;
  const int G = out_size;
  const int P = in_sizes[25] / G;
  const int NCS = in_sizes[10] / (FE * HID);
  const int* srcp = eidx;
  const int* dstp = eidx + E;

  char* wp = (char*)d_ws;
  auto bump = [&](size_t bytes) -> void* {
    void* p = (void*)wp;
    wp += (bytes + 255) & ~((size_t)255);
    return p;
  };
  float* hA    = (float*)bump((size_t)N * HID * 4);
  float* hB    = (float*)bump((size_t)N * HID * 4);
  float* XQ    = (float*)bump((size_t)N * HID * 4);
  float* XV    = (float*)bump((size_t)N * HID * 4);
  float* Ksum  = (float*)bump((size_t)N * DK * 4);
  float* Agg   = (float*)bump((size_t)N * HID * 4);
  float* Zb    = (float*)bump((size_t)N * HID * 4);
  float* EaN   = (float*)bump((size_t)E * FE * 4);
  float* Ve    = (float*)bump((size_t)E * HID * 4);
  float* En    = (float*)bump((size_t)E * HEAD * 4);
  float* We0p  = (float*)bump((size_t)FE * HID * 4);
  float* Wq0p  = (float*)bump((size_t)HID * HID * 4);
  float* Wv0p  = (float*)bump((size_t)HID * HID * 4);
  float* Wkf   = (float*)bump((size_t)HID * DK * 4);
  int*   Coff  = (int*)bump((size_t)(N + 1) * 4);
  int*   Ceid  = (int*)bump((size_t)E * 4);
  int*   Deg   = (int*)bump((size_t)N * 4);
  int*   Cur   = (int*)bump((size_t)N * 4);

  const int T = 256;
  // --- prep ---
  k_h0<<<(N + T - 1) / T, T, 0, stream>>>(x, hA, N);
  k_ea<<<(E + T - 1) / T, T, 0, stream>>>(edge_attr, EaN, E);
  k_pad_we<<<(FE * HID + T - 1) / T, T, 0, stream>>>(c0_We, We0p);
  k_pad_w128<<<(HID * HID + T - 1) / T, T, 0, stream>>>(c0_Wq, Wq0p);
  k_pad_w128<<<(HID * HID + T - 1) / T, T, 0, stream>>>(c0_Wv, Wv0p);
  // --- CSR over dst (once, reused by all layers) ---
  k_zero_i<<<(N + T - 1) / T, T, 0, stream>>>(Deg, N);
  k_zero_i<<<(N + T - 1) / T, T, 0, stream>>>(Cur, N);
  k_count<<<(E + T - 1) / T, T, 0, stream>>>(dstp, Deg, E);
  k_scan<<<1, 1, 0, stream>>>(Deg, Coff, N);
  k_fill<<<(E + T - 1) / T, T, 0, stream>>>(dstp, Coff, Cur, Ceid, E);
  k_sort_csr<<<(N + T - 1) / T, T, 0, stream>>>(Coff, Ceid, N);

  float* hin = hA;
  float* hout = hB;
  for (int L = 0; L < 1 + NCS; ++L) {
    const float *We, *Wq, *Wv, *WkSrc, *Wo, *bo, *gg, *bb;
    int Din;
    if (L == 0) {
      We = We0p; Wq = Wq0p; Wv = Wv0p; WkSrc = c0_Wk; Din = FN;
      Wo = c0_Wo; bo = c0_bo; gg = c0_g; bb = c0_b;
    } else {
      int i = L - 1;
      We = cs_We + (size_t)i * FE * HID;
      Wq = cs_Wq + (size_t)i * HID * HID;
      Wv = cs_Wv + (size_t)i * HID * HID;
      WkSrc = cs_Wk + (size_t)i * HID * HID;
      Din = HID;
      Wo = cs_Wo + (size_t)i * HID * HID;
      bo = cs_bo + (size_t)i * HID;
      gg = cs_g + (size_t)i * HID;
      bb = cs_b + (size_t)i * HID;
    }
    k_fold_wk<<<(HID * DK + T - 1) / T, T, 0, stream>>>(WkSrc, Wkf, Din);
    dim3 gfull((N + 15) / 16, HID / 16);
    k_gemm_node<<<gfull, 32, 0, stream>>>(hin, Wq, nullptr, XQ, N, HID, HID);
    k_gemm_node<<<gfull, 32, 0, stream>>>(hin, Wv, nullptr, XV, N, HID, HID);
    dim3 gks((N + 15) / 16, DK / 16);
    k_gemm_node<<<gks, 32, 0, stream>>>(hin, Wkf, nullptr, Ksum, N, HID, DK);
    k_edge<<<(E + 15) / 16, 256, 0, stream>>>(EaN, srcp, dstp, We, Wq, Wv, XQ, XV,
                                              Ksum, Ve, En, E);
    k_agg<<<N, HID, 0, stream>>>(XQ, XV, Ksum, En, Ve, Coff, Ceid, Agg, N);
    k_gemm_node<<<gfull, 32, 0, stream>>>(Agg, Wo, bo, Zb, N, HID, HID);
    k_ln_tanh<<<N, HID, 0, stream>>>(Zb, gg, bb, hout, N);
    float* tmp = hin; hin = hout; hout = tmp;
  }
  // --- pooling + head ---
  k_pool<<<G, HID, 0, stream>>>(hin, ipos, gate, W1, b1, W2, b2, (float*)d_out, P);
}